// SimHHNode_3676492005584
// MI455X (gfx1250) — compile-verified
//
#include <hip/hip_runtime.h>
#include <hip/hip_bf16.h>
#include <stdint.h>

// Hodgkin–Huxley scan, T=8 steps, 4.19M independent neurons.
// One thread = 4 consecutive neurons (float4); state (mem,m,n,h) in VGPRs.
// Inputs staged global->LDS via gfx1250 async-to-LDS B128 loads (ASYNCcnt),
// waits pipelined so step t's compute overlaps loads of steps t+1..7.
// Trans-op minimized: 3x v_exp_f32 + 3x v_rcp_f32 per neuron-step
// (exp(-(v+c)/10) family shares one exp; exp(-v/20) = exp(-v/80)^4).

#define DT      0.01f
#define DT_CM   0.5f      // DT / CM, CM = 0.02
#define G_L     0.3f
#define V_L     10.6f
#define LOG2E   1.4426950408889634f

#define EXP_M1   0.36787944117144233f   // e^-1
#define EXP_M2P5 0.0820849986238988f    // e^-2.5
#define EXP_M3   0.049787068367863944f  // e^-3

__device__ __forceinline__ float exp2_raw(float x) {
    return __builtin_amdgcn_exp2f(x);   // v_exp_f32
}
__device__ __forceinline__ float fast_rcp(float x) {
    return __builtin_amdgcn_rcpf(x);    // v_rcp_f32
}

__global__ __launch_bounds__(256) void hh_scan_kernel(
    const float* __restrict__ x,      // [8, BC] flattened
    const float* __restrict__ thr_p,  // scalar
    float* __restrict__ out,          // [8, BC]
    int BC)                           // B*C = neurons
{
    // LDS staging: 8 timesteps x 256 threads x 16B = 32 KB
    __shared__ float s_x[8 * 1024];

    const int tid  = threadIdx.x;
    const int base = (blockIdx.x * 256 + tid) * 4;   // first neuron of this thread
    if (base >= BC) return;

    // ---- issue all 8 async global->LDS B128 loads up front ----
    const uint64_t gbase   = (uint64_t)(uintptr_t)x + 4ull * (uint64_t)base;
    const uint32_t lds_row = (uint32_t)(uintptr_t)(&s_x[0]) + (uint32_t)tid * 16u;
#pragma unroll
    for (int t = 0; t < 8; ++t) {
        uint64_t ga = gbase + 4ull * (uint64_t)t * (uint64_t)BC;
        uint32_t la = lds_row + (uint32_t)t * 4096u;
        asm volatile("global_load_async_to_lds_b128 %0, %1, off"
                     :: "v"(la), "v"(ga) : "memory");
    }

    const float thr = *thr_p;

    float mem[4], m[4], n[4], h[4];
#pragma unroll
    for (int j = 0; j < 4; ++j) { mem[j] = 0.f; m[j] = 0.f; n[j] = 0.f; h[j] = 0.f; }

    float* op = out + base;

#pragma unroll
    for (int t = 0; t < 8; ++t) {
        // async loads retire in order: tile t ready once ASYNCcnt <= 7-t.
        asm volatile("s_wait_asynccnt %0" :: "i"(7 - t) : "memory");

        const float4 xv = *reinterpret_cast<const float4*>(&s_x[t * 1024 + tid * 4]);
        const float xs[4] = { xv.x, xv.y, xv.z, xv.w };
        float sp[4];

#pragma unroll
        for (int j = 0; j < 4; ++j) {
            float v  = mem[j];
            // ionic currents
            float m3   = m[j] * m[j] * m[j];
            float I_Na = m3 * h[j] * 120.0f * (v - 120.0f);   // G_NA*(v - V_NA)
            float n2   = n[j] * n[j];
            float I_K  = n2 * n2 * 120.0f * (v + 120.0f);     // G_K*(v - V_K), V_K=-120
            float I_L  = G_L * (v - V_L);
            v = v + DT_CM * (xs[j] - I_Na - I_K - I_L);

            // ---- shared-exponential gating rates ----
            // E10 = exp(-v/10); E80 = exp(-v/80); E18 = exp(-v/18)
            float E10 = exp2_raw(v * (-0.1f * LOG2E));
            float E80 = exp2_raw(v * (-LOG2E / 80.0f));
            float E18 = exp2_raw(v * (-LOG2E / 18.0f));

            float a_n = 0.01f * (v + 10.0f) * fast_rcp(1.0f - E10 * EXP_M1);
            float a_m = 0.1f  * (v + 25.0f) * fast_rcp(1.0f - E10 * EXP_M2P5);
            float b_h = fast_rcp(1.0f + E10 * EXP_M3);

            float b_n = 0.125f * E80;
            float E80sq = E80 * E80;
            float a_h = 0.07f * (E80sq * E80sq);              // exp(-v/20)
            float b_m = 4.0f  * E18;

            // g' = g*(1 - DT*(a+b)) + DT*a   (add + 2 FMA + mul per gate)
            n[j] = __builtin_fmaf(n[j], __builtin_fmaf(-DT, a_n + b_n, 1.0f), DT * a_n);
            m[j] = __builtin_fmaf(m[j], __builtin_fmaf(-DT, a_m + b_m, 1.0f), DT * a_m);
            h[j] = __builtin_fmaf(h[j], __builtin_fmaf(-DT, a_h + b_h, 1.0f), DT * a_h);

            const bool fired = (v - thr) > 0.0f;  // atan surrogate fwd = step
            sp[j]  = fired ? 1.0f : 0.0f;
            mem[j] = fired ? 0.0f : v;            // hard reset on spike
        }

        float4 o;
        o.x = sp[0]; o.y = sp[1]; o.z = sp[2]; o.w = sp[3];
        *reinterpret_cast<float4*>(op + (size_t)t * (size_t)BC) = o;
    }
}

extern "C" void kernel_launch(void* const* d_in, const int* in_sizes, int n_in,
                              void* d_out, int out_size, void* d_ws, size_t ws_size,
                              hipStream_t stream) {
    const float* inputs = (const float*)d_in[0];   // [T=8, B, C] f32
    const float* thr    = (const float*)d_in[1];   // scalar f32
    float* out          = (float*)d_out;           // [T=8, B, C] f32

    const int total = in_sizes[0];     // T*B*C
    const int BC    = total / 8;       // neurons
    const int nthreads = (BC + 3) / 4; // one thread per 4 neurons
    const int block = 256;
    const int grid  = (nthreads + block - 1) / block;

    hh_scan_kernel<<<grid, block, 0, stream>>>(inputs, thr, out, BC);
}